// SimpleLESS4FDModel_88734024335900
// MI455X (gfx1250) — compile-verified
//
#include <hip/hip_runtime.h>

#define NNODES 50000
#define NEDGES 1000000
#define TOTE   (NEDGES + NNODES)   // edges + self loops
#define DIN    768
#define HIDC   128
#define NHD    4
#define CPH    32
#define SLOPE  0.2f
#define LN_EPS 1e-5f

typedef float v2f __attribute__((ext_vector_type(2)));
typedef float v8f __attribute__((ext_vector_type(8)));

// ---------------------------------------------------------------------------
// Dense GEMM: C[M x Nc] = A[M x K] @ W[K x Nc] + bias, optional ReLU.
// One wave computes a 32x64 strip: 2 M-subtiles x 4 N-subtiles, so each B
// fragment feeds 2 WMMAs (8 WMMA : 10 VMEM per K-step).
// Fragment layouts per CDNA5 ISA 7.12.2 (V_WMMA_F32_16X16X4_F32):
//   A 16x4 : lanes 0-15 hold K={0,1}, lanes 16-31 hold K={2,3} (2 VGPRs)
//   B 4x16 : VGPR0 = rows K0|K2, VGPR1 = rows K1|K3, N striped over lanes
//   C 16x16: VGPR r -> row r (lanes 0-15) / row r+8 (lanes 16-31)
// ---------------------------------------------------------------------------
#define WMMA_F32(Acc, Af, Bf) \
    __builtin_amdgcn_wmma_f32_16x16x4_f32(false, Af, false, Bf, (short)0, Acc, false, false)

__global__ __launch_bounds__(160)
void gemm_bias_wmma(const float* __restrict__ A, const float* __restrict__ W,
                    const float* __restrict__ bias, float* __restrict__ C,
                    int M, int K, int Nc, int relu)
{
    const int wave  = threadIdx.x >> 5;
    const int lane  = threadIdx.x & 31;
    const int mtile = blockIdx.x * 5 + wave;
    const int m0    = mtile * 32;
    if (m0 >= M) return;                     // wave-uniform: EXEC stays all-ones
    const int n0 = blockIdx.y * 64;
    const int hh = lane >> 4;                // lane half: 0 -> K{0,1}, 1 -> K{2,3}
    const int mr = lane & 15;
    const bool hasM2 = (m0 + 16) < M;        // wave-uniform tail guard
    const int r0 = m0 + mr;
    const int r1 = hasM2 ? (m0 + 16 + mr) : r0;   // clamp reads on tail tile
    const float* arow0 = A + (size_t)r0 * K;
    const float* arow1 = A + (size_t)r1 * K;

    v8f c00 = {}, c01 = {}, c02 = {}, c03 = {};
    v8f c10 = {}, c11 = {}, c12 = {}, c13 = {};

    for (int k = 0; k < K; k += 4) {
        v2f a0 = *(const v2f*)(arow0 + k + hh * 2);   // 8B-aligned
        v2f a1 = *(const v2f*)(arow1 + k + hh * 2);
        const float* b0p = W + (size_t)(k + hh * 2) * Nc + n0 + mr;
        const float* b1p = b0p + Nc;
        v2f b0, b1, b2, b3;
        b0.x = b0p[0];  b0.y = b1p[0];
        b1.x = b0p[16]; b1.y = b1p[16];
        b2.x = b0p[32]; b2.y = b1p[32];
        b3.x = b0p[48]; b3.y = b1p[48];
        c00 = WMMA_F32(c00, a0, b0);  c10 = WMMA_F32(c10, a1, b0);
        c01 = WMMA_F32(c01, a0, b1);  c11 = WMMA_F32(c11, a1, b1);
        c02 = WMMA_F32(c02, a0, b2);  c12 = WMMA_F32(c12, a1, b2);
        c03 = WMMA_F32(c03, a0, b3);  c13 = WMMA_F32(c13, a1, b3);
    }

    v8f acc[8] = {c00, c01, c02, c03, c10, c11, c12, c13};
    #pragma unroll
    for (int g = 0; g < 2; ++g) {
        if (g == 1 && !hasM2) break;         // wave-uniform
        const int mb = m0 + g * 16;
        #pragma unroll
        for (int t = 0; t < 4; ++t) {
            const int col = n0 + t * 16 + mr;
            const float bv = bias[col];
            #pragma unroll
            for (int r = 0; r < 8; ++r) {
                float v = acc[g * 4 + t][r] + bv;
                if (relu) v = fmaxf(v, 0.f);
                C[(size_t)(mb + r + hh * 8) * Nc + col] = v;
            }
        }
    }
}

// ------------------------- CSR build (dst-major, built once) ----------------
__global__ void csr_zero(int* __restrict__ deg)
{
    int i = blockIdx.x * blockDim.x + threadIdx.x;
    if (i < NNODES) deg[i] = 0;
}

__global__ void csr_count(const int* __restrict__ ei, int* __restrict__ deg)
{
    int e = blockIdx.x * blockDim.x + threadIdx.x;
    if (e >= TOTE) return;
    int d = (e < NEDGES) ? ei[NEDGES + e] : e - NEDGES;
    atomicAdd(&deg[d], 1);
}

// Single-block exclusive scan of deg[0..N) -> rowstart[0..N], copy into pos.
__global__ __launch_bounds__(1024)
void csr_scan(const int* __restrict__ deg, int* __restrict__ rowstart,
              int* __restrict__ pos)
{
    __shared__ int buf[1024];
    __shared__ int carry;
    if (threadIdx.x == 0) carry = 0;
    __syncthreads();
    for (int base = 0; base < NNODES; base += 1024) {
        const int i = base + (int)threadIdx.x;
        const int v = (i < NNODES) ? deg[i] : 0;
        buf[threadIdx.x] = v;
        __syncthreads();
        for (int off = 1; off < 1024; off <<= 1) {
            int t = (threadIdx.x >= (unsigned)off) ? buf[threadIdx.x - off] : 0;
            __syncthreads();
            buf[threadIdx.x] += t;
            __syncthreads();
        }
        const int incl = buf[threadIdx.x];
        if (i < NNODES) {
            int rs = carry + incl - v;       // exclusive
            rowstart[i] = rs;
            pos[i] = rs;
        }
        __syncthreads();
        if (threadIdx.x == 1023) carry += incl;
        __syncthreads();
    }
    if (threadIdx.x == 0) rowstart[NNODES] = carry;   // == TOTE
}

__global__ void csr_fill(const int* __restrict__ ei, int* __restrict__ pos,
                         int* __restrict__ csr_src, int* __restrict__ csr_eid)
{
    int e = blockIdx.x * blockDim.x + threadIdx.x;
    if (e >= TOTE) return;
    int s, d;
    if (e < NEDGES) { s = ei[e]; d = ei[NEDGES + e]; } else { s = d = e - NEDGES; }
    int idx = atomicAdd(&pos[d], 1);
    csr_src[idx] = s;
    csr_eid[idx] = e;
}

// ------------------------- GATv2 edge score (edge-parallel) ----------------
// score[e,h] = att[h] . leaky_relu(xl[src] + xr[dst]); stored as float4/edge.
__global__ void edge_score(const int* __restrict__ ei, const float* __restrict__ xl,
                           const float* __restrict__ xr, const float* __restrict__ att,
                           float* __restrict__ score)
{
    int tid = blockIdx.x * blockDim.x + threadIdx.x;
    if (tid >= TOTE * NHD) return;
    const int e = tid >> 2, hd = tid & 3;
    int s, d;
    if (e < NEDGES) { s = ei[e]; d = ei[NEDGES + e]; } else { s = d = e - NEDGES; }
    const float4* pl = (const float4*)(xl + (size_t)s * HIDC + hd * CPH);
    const float4* pr = (const float4*)(xr + (size_t)d * HIDC + hd * CPH);
    const float4* pa = (const float4*)(att + hd * CPH);
    float sc = 0.f;
    #pragma unroll
    for (int j = 0; j < 8; ++j) {
        float4 a = pl[j], b = pr[j], w = pa[j];
        float v;
        v = a.x + b.x; v = v > 0.f ? v : SLOPE * v; sc += w.x * v;
        v = a.y + b.y; v = v > 0.f ? v : SLOPE * v; sc += w.y * v;
        v = a.z + b.z; v = v > 0.f ? v : SLOPE * v; sc += w.z * v;
        v = a.w + b.w; v = v > 0.f ? v : SLOPE * v; sc += w.w * v;
    }
    score[tid] = sc;
}

// ------------------- GATv2 aggregation: one wave per dst node --------------
// Atomic-free: softmax stats via wave shuffles; phase C reads each source row
// as one fully-coalesced 512B wave transaction (32 lanes x float4) from L2.
__global__ void gat_gather(const int* __restrict__ rowstart,
                           const int* __restrict__ csr_src,
                           const int* __restrict__ csr_eid,
                           const float* __restrict__ score,
                           const float* __restrict__ xl,
                           const float* __restrict__ conv_b,
                           float* __restrict__ h)
{
    const int wave = threadIdx.x >> 5, lane = threadIdx.x & 31;
    const int i = blockIdx.x * 8 + wave;
    if (i >= NNODES) return;
    const int beg = rowstart[i], end = rowstart[i + 1];   // >= 1 edge (self loop)

    // Phase A: per-head max
    float mx0 = -1e30f, mx1 = -1e30f, mx2 = -1e30f, mx3 = -1e30f;
    for (int j = beg + lane; j < end; j += 32) {
        const float4 sc = *(const float4*)(score + (size_t)csr_eid[j] * 4);
        mx0 = fmaxf(mx0, sc.x); mx1 = fmaxf(mx1, sc.y);
        mx2 = fmaxf(mx2, sc.z); mx3 = fmaxf(mx3, sc.w);
    }
    #pragma unroll
    for (int off = 16; off; off >>= 1) {
        mx0 = fmaxf(mx0, __shfl_xor(mx0, off, 32));
        mx1 = fmaxf(mx1, __shfl_xor(mx1, off, 32));
        mx2 = fmaxf(mx2, __shfl_xor(mx2, off, 32));
        mx3 = fmaxf(mx3, __shfl_xor(mx3, off, 32));
    }

    // Phase B: per-head denom
    float dn0 = 0.f, dn1 = 0.f, dn2 = 0.f, dn3 = 0.f;
    for (int j = beg + lane; j < end; j += 32) {
        const float4 sc = *(const float4*)(score + (size_t)csr_eid[j] * 4);
        dn0 += expf(sc.x - mx0); dn1 += expf(sc.y - mx1);
        dn2 += expf(sc.z - mx2); dn3 += expf(sc.w - mx3);
    }
    #pragma unroll
    for (int off = 16; off; off >>= 1) {
        dn0 += __shfl_xor(dn0, off, 32);
        dn1 += __shfl_xor(dn1, off, 32);
        dn2 += __shfl_xor(dn2, off, 32);
        dn3 += __shfl_xor(dn3, off, 32);
    }

    // Phase C: lane owns columns [lane*4, lane*4+4) -> head = lane>>3
    const int hd = lane >> 3;
    const float m_h = (hd < 2) ? (hd == 0 ? mx0 : mx1) : (hd == 2 ? mx2 : mx3);
    const float d_h = (hd < 2) ? (hd == 0 ? dn0 : dn1) : (hd == 2 ? dn2 : dn3);
    float a0 = 0.f, a1 = 0.f, a2 = 0.f, a3 = 0.f;
    for (int j = beg; j < end; ++j) {                 // uniform across wave
        const int s   = csr_src[j];                   // broadcast load
        const int eid = csr_eid[j];
        const float p = expf(score[(size_t)eid * 4 + hd] - m_h);
        const float4 v = *(const float4*)(xl + (size_t)s * HIDC + lane * 4);
        a0 += p * v.x; a1 += p * v.y; a2 += p * v.z; a3 += p * v.w;
    }
    const float inv = 1.f / d_h;
    const float4 cb = *(const float4*)(conv_b + lane * 4);
    float4 o;
    o.x = fmaxf(a0 * inv + cb.x, 0.f);
    o.y = fmaxf(a1 * inv + cb.y, 0.f);
    o.z = fmaxf(a2 * inv + cb.z, 0.f);
    o.w = fmaxf(a3 * inv + cb.w, 0.f);
    *(float4*)(h + (size_t)i * HIDC + lane * 4) = o;  // relu(out + conv_b)
}

// ----------------------- residual + LayerNorm (wave/row, in-place ok) ------
__global__ void residual_ln(const float* h, const float* __restrict__ a2,
                            const float* __restrict__ g, const float* __restrict__ b,
                            float* zn)
{
    const int wave = threadIdx.x >> 5, lane = threadIdx.x & 31;
    const int row = blockIdx.x * 8 + wave;
    if (row >= NNODES) return;
    float4 hv = ((const float4*)(h + (size_t)row * HIDC))[lane];
    float4 av = ((const float4*)(a2 + (size_t)row * HIDC))[lane];
    float4 z;
    z.x = hv.x + av.x; z.y = hv.y + av.y; z.z = hv.z + av.z; z.w = hv.w + av.w;
    float s = z.x + z.y + z.z + z.w;
    #pragma unroll
    for (int off = 16; off; off >>= 1) s += __shfl_xor(s, off, 32);
    const float mu = s * (1.f / HIDC);
    float dx = z.x - mu, dy = z.y - mu, dz = z.z - mu, dw = z.w - mu;
    float ss = dx * dx + dy * dy + dz * dz + dw * dw;
    #pragma unroll
    for (int off = 16; off; off >>= 1) ss += __shfl_xor(ss, off, 32);
    const float inv = rsqrtf(ss * (1.f / HIDC) + LN_EPS);
    float4 gv = ((const float4*)g)[lane];
    float4 bv = ((const float4*)b)[lane];
    float4 o;
    o.x = dx * inv * gv.x + bv.x;
    o.y = dy * inv * gv.y + bv.y;
    o.z = dz * inv * gv.z + bv.z;
    o.w = dw * inv * gv.w + bv.w;
    ((float4*)(zn + (size_t)row * HIDC))[lane] = o;
}

// ------------------------- final 64 -> 2 classifier ------------------------
__global__ void classifier2(const float* __restrict__ c1, const float* __restrict__ W2,
                            const float* __restrict__ b2, float* __restrict__ out)
{
    int i = blockIdx.x * blockDim.x + threadIdx.x;
    if (i >= NNODES) return;
    const float* r = c1 + (size_t)i * 64;
    float o0 = b2[0], o1 = b2[1];
    #pragma unroll
    for (int k = 0; k < 64; ++k) {
        float v = r[k];
        o0 += v * W2[2 * k + 0];
        o1 += v * W2[2 * k + 1];
    }
    out[2 * i + 0] = o0;
    out[2 * i + 1] = o1;
}

// ---------------------------------------------------------------------------
extern "C" void kernel_launch(void* const* d_in, const int* in_sizes, int n_in,
                              void* d_out, int out_size, void* d_ws, size_t ws_size,
                              hipStream_t stream)
{
    (void)in_sizes; (void)n_in; (void)out_size; (void)ws_size;
    const float* x      = (const float*)d_in[0];
    const int*   ei     = (const int*)  d_in[1];
    const float* emb_W  = (const float*)d_in[2];
    const float* emb_b  = (const float*)d_in[3];
    const float* Wl     = (const float*)d_in[4];
    const float* bl     = (const float*)d_in[5];
    const float* Wr     = (const float*)d_in[6];
    const float* br     = (const float*)d_in[7];
    const float* att    = (const float*)d_in[8];
    const float* conv_b = (const float*)d_in[9];
    const float* Wv     = (const float*)d_in[10];
    const float* bvv    = (const float*)d_in[11];
    const float* Wo     = (const float*)d_in[12];
    const float* bo     = (const float*)d_in[13];
    const float* ln_g   = (const float*)d_in[14];
    const float* ln_b   = (const float*)d_in[15];
    const float* cW1    = (const float*)d_in[16];
    const float* cb1    = (const float*)d_in[17];
    const float* cW2    = (const float*)d_in[18];
    const float* cb2    = (const float*)d_in[19];

    // workspace: h | xl | xr | score | rowstart | pos | deg | csr_src | csr_eid
    float* ws = (float*)d_ws;
    const size_t SZ = (size_t)NNODES * HIDC;            // 6.4M floats
    float* h     = ws;
    float* xl    = ws + SZ;
    float* xr    = ws + 2 * SZ;
    float* score = ws + 3 * SZ;                         // TOTE*4 = 4.2M floats
    int* rowstart = (int*)(score + (size_t)TOTE * NHD); // N+1
    int* pos      = rowstart + (NNODES + 1);
    int* deg      = pos + NNODES;
    int* csr_src  = deg + NNODES;                       // TOTE
    int* csr_eid  = csr_src + TOTE;                     // TOTE
    // total ~25.7M elements (~103 MB)

    const dim3 gBlk(160);
    const dim3 gGrid(313, 2);                           // ceil(1563/5) M-tiles
    const int  eBlks  = (TOTE * NHD + 255) / 256;
    const int  tBlks  = (TOTE + 255) / 256;
    const int  nBlks  = (NNODES + 7) / 8;

    // --- CSR by destination (shared by both layers) ---
    csr_zero <<<(NNODES + 255) / 256, 256, 0, stream>>>(deg);
    csr_count<<<tBlks, 256, 0, stream>>>(ei, deg);
    csr_scan <<<1, 1024, 0, stream>>>(deg, rowstart, pos);
    csr_fill <<<tBlks, 256, 0, stream>>>(ei, pos, csr_src, csr_eid);

    // --- node embedding: h = x @ emb_W + emb_b (K = 768) ---
    gemm_bias_wmma<<<gGrid, gBlk, 0, stream>>>(x, emb_W, emb_b, h, NNODES, DIN, HIDC, 0);

    for (int l = 0; l < 2; ++l) {
        gemm_bias_wmma<<<gGrid, gBlk, 0, stream>>>(h, Wl + (size_t)l * HIDC * HIDC,
                                                   bl + l * HIDC, xl, NNODES, HIDC, HIDC, 0);
        gemm_bias_wmma<<<gGrid, gBlk, 0, stream>>>(h, Wr + (size_t)l * HIDC * HIDC,
                                                   br + l * HIDC, xr, NNODES, HIDC, HIDC, 0);
        edge_score<<<eBlks, 256, 0, stream>>>(ei, xl, xr, att + l * NHD * CPH, score);
        gat_gather<<<nBlks, 256, 0, stream>>>(rowstart, csr_src, csr_eid, score, xl,
                                              conv_b + l * HIDC, h);
    }

    // entity attention over seq_len=1: attended = (h@Wv+bv)@Wo+bo
    gemm_bias_wmma<<<gGrid, gBlk, 0, stream>>>(h, Wv, bvv, xl, NNODES, HIDC, HIDC, 0);
    gemm_bias_wmma<<<gGrid, gBlk, 0, stream>>>(xl, Wo, bo, xr, NNODES, HIDC, HIDC, 0);
    // z = h + attended; LayerNorm (in-place into h)
    residual_ln<<<nBlks, 256, 0, stream>>>(h, xr, ln_g, ln_b, h);
    // classifier: c1 = relu(zn @ cW1 + cb1) -> xl ; logits -> d_out
    gemm_bias_wmma<<<dim3(313, 1), gBlk, 0, stream>>>(h, cW1, cb1, xl, NNODES, HIDC, 64, 1);
    classifier2<<<(NNODES + 255) / 256, 256, 0, stream>>>(xl, cW2, cb2, (float*)d_out);
}